// pointcnn_79714593014268
// MI455X (gfx1250) — compile-verified
//
#include <hip/hip_runtime.h>

#define NPTS 4096
#define KNN 16
#define NB   8
#define CNTF 524288.0f   // B*N*K
#define LDW  68          // padded LDS row stride (floats)
#define PPW  8           // points per wave in mlp kernels
#define PPB  64          // points per block (8 waves)
#define MLPBLKS 512      // 32768 / PPB

typedef float v2f __attribute__((ext_vector_type(2)));
typedef float v8f __attribute__((ext_vector_type(8)));

// ---------------------------------------------------------------------------
// Kernel 1: per-batch KNN (17 farthest, drop first) + rel first/second moments
// Points packed as float4 {x,y,z,|p|^2} in LDS -> 1x ds_load_b128 per candidate
// ---------------------------------------------------------------------------
__global__ __launch_bounds__(256) void knn_kernel(const float* __restrict__ xyz,
                                                  int* __restrict__ idx_out,
                                                  float* __restrict__ relpart) {
    __shared__ float4 pts[NPTS];           // 64 KB
    const int tid = threadIdx.x;
    const int b   = blockIdx.y;
    const float* xb = xyz + (size_t)b * 3 * NPTS;
    for (int j = tid; j < NPTS; j += 256) {
        const float x = xb[j];
        const float y = xb[NPTS + j];
        const float z = xb[2 * NPTS + j];
        pts[j] = make_float4(x, y, z, x * x + y * y + z * z);
    }
    __syncthreads();

    const int i = blockIdx.x * 256 + tid;
    const float4 me = pts[i];
    const float mx = me.x, my = me.y, mz = me.z, msq = me.w;

    float bd[KNN + 1];
    int   bi[KNN + 1];
#pragma unroll
    for (int t = 0; t <= KNN; ++t) { bd[t] = -__builtin_inff(); bi[t] = 0; }

#pragma unroll 4
    for (int j = 0; j < NPTS; ++j) {
        const float4 q = pts[j];
        const float d = msq + q.w - 2.0f * (mx * q.x + my * q.y + mz * q.z);
        if (d > bd[KNN]) {           // strictly greater: stable tie order == top_k
            bd[KNN] = d; bi[KNN] = j;
#pragma unroll
            for (int t = KNN; t > 0; --t) {
                if (bd[t] > bd[t - 1]) {
                    float td = bd[t]; bd[t] = bd[t - 1]; bd[t - 1] = td;
                    int   ti = bi[t]; bi[t] = bi[t - 1]; bi[t - 1] = ti;
                }
            }
        }
    }

    float s[9];
#pragma unroll
    for (int c = 0; c < 9; ++c) s[c] = 0.0f;
#pragma unroll
    for (int k = 1; k <= KNN; ++k) {     // drop rank 0 (the single farthest)
        const int j = bi[k];
        idx_out[(size_t)(b * NPTS + i) * KNN + (k - 1)] = j;
        const float4 q = pts[j];
        const float rx = q.x - mx, ry = q.y - my, rz = q.z - mz;
        s[0] += rx;      s[1] += ry;      s[2] += rz;
        s[3] += rx * rx; s[4] += ry * ry; s[5] += rz * rz;
        s[6] += rx * ry; s[7] += rx * rz; s[8] += ry * rz;
    }

    __syncthreads();                     // done reading point LDS; reuse as scratch
    float* red = (float*)pts;            // need 256*9 floats << 16384
#pragma unroll
    for (int c = 0; c < 9; ++c) red[c * 256 + tid] = s[c];
    __syncthreads();
    for (int st = 128; st > 0; st >>= 1) {
        if (tid < st) {
#pragma unroll
            for (int c = 0; c < 9; ++c) red[c * 256 + tid] += red[c * 256 + tid + st];
        }
        __syncthreads();
    }
    if (tid < 9) {
        const int bid = b * 16 + blockIdx.x;
        relpart[bid * 16 + tid] = red[tid * 256];
    }
}

// ---------------------------------------------------------------------------
// Kernel 2: BN0 stats analytically from rel moments -> scale/shift (s0,t0)
// ---------------------------------------------------------------------------
__global__ void stats0_kernel(const float* __restrict__ relpart,
                              const float* __restrict__ w1, const float* __restrict__ b1,
                              const float* __restrict__ gamma, const float* __restrict__ beta,
                              float* __restrict__ s0t0) {
    __shared__ float S[9];
    const int tid = threadIdx.x;
    if (tid < 9) {
        float a = 0.0f;
        for (int blk = 0; blk < 128; ++blk) a += relpart[blk * 16 + tid];
        S[tid] = a;
    }
    __syncthreads();
    if (tid < 64) {
        const float inv = 1.0f / CNTF;
        const float m0 = S[0] * inv, m1 = S[1] * inv, m2 = S[2] * inv;
        const float Mxx = S[3] * inv, Myy = S[4] * inv, Mzz = S[5] * inv;
        const float Mxy = S[6] * inv, Mxz = S[7] * inv, Myz = S[8] * inv;
        const float wx = w1[tid * 3], wy = w1[tid * 3 + 1], wz = w1[tid * 3 + 2];
        const float bb = b1[tid];
        const float mul = wx * m0 + wy * m1 + wz * m2;
        const float mu  = mul + bb;
        const float e2  = wx * wx * Mxx + wy * wy * Myy + wz * wz * Mzz
                        + 2.0f * (wx * wy * Mxy + wx * wz * Mxz + wy * wz * Myz)
                        + 2.0f * bb * mul + bb * bb;
        const float var  = e2 - mu * mu;
        const float istd = rsqrtf(var + 1e-5f);
        const float sc   = gamma[tid] * istd;
        s0t0[tid]      = sc;
        s0t0[64 + tid] = beta[tid] - mu * sc;
    }
}

// ---------------------------------------------------------------------------
// Kernels 3 & 5: WMMA MLP pass. FINAL=false -> accumulate BN1 stats of h1.
//               FINAL=true  -> a1 = relu(bn1(h1)); h2 = w2@a1+b2; max over K.
// One wave handles one point per step: its 16 neighbors = one 16-col tile.
// ---------------------------------------------------------------------------
template <bool FINAL>
__global__ __launch_bounds__(256) void mlp_kernel(
        const float* __restrict__ xyz, const int* __restrict__ idx,
        const float* __restrict__ w1, const float* __restrict__ b1,
        const float* __restrict__ w2, const float* __restrict__ b2,
        const float* __restrict__ s0t0, const float* __restrict__ s1t1,
        float* __restrict__ h1part, float* __restrict__ out) {
    __shared__ float W2s[64 * LDW];
    __shared__ float A0t[8][16 * LDW];     // per-wave column-major activation tile
    __shared__ float w1s[64 * 4];
    __shared__ float b1c[64], b2s[64], s0s[64], t0s[64], s1s[64], t1s[64];

    const int tid  = threadIdx.x;
    const int wave = tid >> 5;
    const int lane = tid & 31;
    const int col  = lane & 15;
    const int half = lane >> 4;

    for (int r = tid; r < 64 * 64; r += 256)
        W2s[(r >> 6) * LDW + (r & 63)] = w2[r];
    if (tid < 64) {
        w1s[tid * 4]     = w1[tid * 3];
        w1s[tid * 4 + 1] = w1[tid * 3 + 1];
        w1s[tid * 4 + 2] = w1[tid * 3 + 2];
        w1s[tid * 4 + 3] = 0.0f;
        b1c[tid] = b1[tid];
        b2s[tid] = b2[tid];
        s0s[tid] = s0t0[tid];
        t0s[tid] = s0t0[64 + tid];
        if (FINAL) { s1s[tid] = s1t1[tid]; t1s[tid] = s1t1[64 + tid]; }
        else       { s1s[tid] = 0.0f;      t1s[tid] = 0.0f; }
    }
    __syncthreads();

    float* myA = A0t[wave];
    float sumv[4][8], sumq[4][8];
#pragma unroll
    for (int m = 0; m < 4; ++m)
#pragma unroll
        for (int r = 0; r < 8; ++r) { sumv[m][r] = 0.0f; sumq[m][r] = 0.0f; }

    for (int t = 0; t < PPW; ++t) {
        const int p = blockIdx.x * PPB + wave * PPW + t;
        const int b = p >> 12;
        const int n = p & (NPTS - 1);
        const float* xb = xyz + (size_t)b * 3 * NPTS;
        const float cx = xb[n], cy = xb[NPTS + n], cz = xb[2 * NPTS + n];
        const int j  = idx[(size_t)p * KNN + col];
        const float rx = xb[j] - cx, ry = xb[NPTS + j] - cy, rz = xb[2 * NPTS + j] - cz;

        // layer0 + BN0 + relu, write a0 column-major into LDS (bank-tuned)
#pragma unroll
        for (int cc = 0; cc < 32; ++cc) {
            const int c = 2 * cc + half;
            const float h0 = w1s[c * 4] * rx + w1s[c * 4 + 1] * ry + w1s[c * 4 + 2] * rz + b1c[c];
            myA[col * LDW + c] = fmaxf(h0 * s0s[c] + t0s[c], 0.0f);
        }
        __syncthreads();

        // layer1: h1 = W2 @ A0 + b2 via V_WMMA_F32_16X16X4_F32
        v8f acc[4];
#pragma unroll
        for (int m = 0; m < 4; ++m) {
#pragma unroll
            for (int r = 0; r < 8; ++r) acc[m][r] = b2s[16 * m + 8 * half + r];
#pragma unroll
            for (int kk = 0; kk < 16; ++kk) {
                const int k0 = kk * 4 + 2 * half;
                v2f a = *(const v2f*)&W2s[(16 * m + col) * LDW + k0];
                v2f bv = *(const v2f*)&myA[col * LDW + k0];
                acc[m] = __builtin_amdgcn_wmma_f32_16x16x4_f32(
                    false, a, false, bv, (short)0, acc[m], false, false);
            }
        }

        if (!FINAL) {
#pragma unroll
            for (int m = 0; m < 4; ++m)
#pragma unroll
                for (int r = 0; r < 8; ++r) {
                    const float v = acc[m][r];
                    sumv[m][r] += v;
                    sumq[m][r] += v * v;
                }
        } else {
            // a1 = relu(bn1(h1)); overwrite activation tile
#pragma unroll
            for (int m = 0; m < 4; ++m)
#pragma unroll
                for (int r = 0; r < 8; ++r) {
                    const int c = 16 * m + 8 * half + r;
                    myA[col * LDW + c] = fmaxf(acc[m][r] * s1s[c] + t1s[c], 0.0f);
                }
            __syncthreads();

            // layer2: h2 = W2 @ A1 + b2, then max over the 16 columns (K)
            v8f acc2[4];
#pragma unroll
            for (int m = 0; m < 4; ++m) {
#pragma unroll
                for (int r = 0; r < 8; ++r) acc2[m][r] = b2s[16 * m + 8 * half + r];
#pragma unroll
                for (int kk = 0; kk < 16; ++kk) {
                    const int k0 = kk * 4 + 2 * half;
                    v2f a = *(const v2f*)&W2s[(16 * m + col) * LDW + k0];
                    v2f bv = *(const v2f*)&myA[col * LDW + k0];
                    acc2[m] = __builtin_amdgcn_wmma_f32_16x16x4_f32(
                        false, a, false, bv, (short)0, acc2[m], false, false);
                }
            }
#pragma unroll
            for (int m = 0; m < 4; ++m)
#pragma unroll
                for (int r = 0; r < 8; ++r) {
                    float v = acc2[m][r];
                    v = fmaxf(v, __shfl_xor(v, 1, 32));
                    v = fmaxf(v, __shfl_xor(v, 2, 32));
                    v = fmaxf(v, __shfl_xor(v, 4, 32));
                    v = fmaxf(v, __shfl_xor(v, 8, 32));
                    if (col == 0) {
                        const int c = 16 * m + 8 * half + r;
                        out[(((size_t)b * 64 + c) << 12) + n] = v;
                    }
                }
        }
    }

    if (!FINAL) {
        // reduce sums over the 16 columns, then across waves (fixed order)
#pragma unroll
        for (int m = 0; m < 4; ++m)
#pragma unroll
            for (int r = 0; r < 8; ++r) {
                float v = sumv[m][r], q = sumq[m][r];
                v += __shfl_xor(v, 1, 32); q += __shfl_xor(q, 1, 32);
                v += __shfl_xor(v, 2, 32); q += __shfl_xor(q, 2, 32);
                v += __shfl_xor(v, 4, 32); q += __shfl_xor(q, 4, 32);
                v += __shfl_xor(v, 8, 32); q += __shfl_xor(q, 8, 32);
                if (col == 0) {
                    const int c = 16 * m + 8 * half + r;
                    myA[c] = v;
                    myA[64 + c] = q;
                }
            }
        __syncthreads();
        if (tid < 128) {
            float a = 0.0f;
            for (int w = 0; w < 8; ++w) a += A0t[w][tid];
            h1part[blockIdx.x * 128 + tid] = a;
        }
    }
}

// ---------------------------------------------------------------------------
// Kernel 4: BN1 stats from per-block partials -> scale/shift (s1,t1)
// ---------------------------------------------------------------------------
__global__ void stats1_kernel(const float* __restrict__ h1part,
                              const float* __restrict__ gamma, const float* __restrict__ beta,
                              float* __restrict__ s1t1) {
    __shared__ float S[128];
    const int tid = threadIdx.x;
    float a = 0.0f;
    for (int blk = 0; blk < MLPBLKS; ++blk) a += h1part[blk * 128 + tid];
    S[tid] = a;
    __syncthreads();
    if (tid < 64) {
        const float inv = 1.0f / CNTF;
        const float mu   = S[tid] * inv;
        const float var  = S[64 + tid] * inv - mu * mu;
        const float istd = rsqrtf(var + 1e-5f);
        const float sc   = gamma[tid] * istd;
        s1t1[tid]      = sc;
        s1t1[64 + tid] = beta[tid] - mu * sc;
    }
}

// ---------------------------------------------------------------------------
extern "C" void kernel_launch(void* const* d_in, const int* in_sizes, int n_in,
                              void* d_out, int out_size, void* d_ws, size_t ws_size,
                              hipStream_t stream) {
    (void)in_sizes; (void)n_in; (void)out_size; (void)ws_size;
    const float* xyz   = (const float*)d_in[0];
    const float* w1    = (const float*)d_in[1];
    const float* b1    = (const float*)d_in[2];
    const float* w2    = (const float*)d_in[3];
    const float* b2    = (const float*)d_in[4];
    const float* gamma = (const float*)d_in[5];
    const float* beta  = (const float*)d_in[6];
    float* out = (float*)d_out;

    char* ws = (char*)d_ws;
    int*   idx     = (int*)  (ws);                 // 8*4096*16*4   = 2,097,152 B
    float* relpart = (float*)(ws + 2097152);       // 128*16*4      =     8,192 B
    float* s0t0    = (float*)(ws + 2105344);       // 128*4         =       512 B
    float* h1part  = (float*)(ws + 2105856);       // 512*128*4     =   262,144 B
    float* s1t1    = (float*)(ws + 2368000);       // 128*4         =       512 B

    knn_kernel<<<dim3(16, NB), 256, 0, stream>>>(xyz, idx, relpart);
    stats0_kernel<<<1, 64, 0, stream>>>(relpart, w1, b1, gamma, beta, s0t0);
    mlp_kernel<false><<<MLPBLKS, 256, 0, stream>>>(xyz, idx, w1, b1, w2, b2,
                                                   s0t0, s0t0, h1part, out);
    stats1_kernel<<<1, 128, 0, stream>>>(h1part, gamma, beta, s1t1);
    mlp_kernel<true><<<MLPBLKS, 256, 0, stream>>>(xyz, idx, w1, b1, w2, b2,
                                                  s0t0, s1t1, h1part, out);
}